// DistanceEncoderHSTLSTM_23287312679166
// MI455X (gfx1250) — compile-verified
//
#include <hip/hip_runtime.h>
#include <hip/hip_bf16.h>

// DistanceEncoderHSTLSTM on MI455X (gfx1250, wave32).
//
// out[i,:] = hd_i * W[lb_i,:] + ld_i * W[hb_i,:]
// Recast as GEMM: out = C x W, C is N x 16 with <=2 nonzeros per row
// (coefficient hd at col lb, ld at col hb; they sum to 1.0 at lb==hb which
// reproduces the at_end case exactly).
//
// One wave handles a 16-element tile: A = 16x16 coefficient matrix,
// B = W (16x64), D = 16x64 output tile. Done as 4 N-blocks of 16 columns,
// each accumulated over K=16 via 4 chained V_WMMA_F32_16X16X4_F32.
//
// VGPR layouts per CDNA5 ISA 7.12.2:
//  A (16x4 f32):  lanes 0-15: v0=K0, v1=K1 ; lanes 16-31: v0=K2, v1=K3 (M=lane%16)
//  C/D (16x16 f32): vgpr r: lanes 0-15 -> M=r, lanes 16-31 -> M=r+8, N=lane%16
//  B (4x16 f32): assumed mirror of A: v0 = row K0 (lanes 0-15) / K2 (lanes 16-31),
//                v1 = row K1 / K3, N=lane%16.  (Compile-only: not HW-verifiable.)
//
// Memory-bound: ~1.09 GB traffic -> ~47 us floor at 23.3 TB/s.

typedef __attribute__((ext_vector_type(2))) float v2f;
typedef __attribute__((ext_vector_type(8))) float v8f;

#define N_SLOTS 16
#define DIM 64

__global__ __launch_bounds__(256)
void dist_encoder_wmma_kernel(const float* __restrict__ dist,
                              const float* __restrict__ W,      // [16][64]
                              const float* __restrict__ slots,  // [16]
                              float* __restrict__ out,          // [N][64]
                              int n)
{
    const int lane = threadIdx.x & 31;
    const int m    = lane & 15;   // element-in-tile (A rows) / output column-in-block
    const int half = lane >> 4;   // 0: K 0,1 ; 1: K 2,3 within each K-chunk

    // Slot boundaries: pointer+index uniform -> compiler emits s_load, held in SGPRs.
    float s[N_SLOTS];
#pragma unroll
    for (int k = 0; k < N_SLOTS; ++k) s[k] = slots[k];

    // Preload all B fragments of W once per wave (4 KB table, L0/L2 resident).
    // bfrag[j][nb]: K-chunk j (rows 4j..4j+3), N-block nb (cols 16*nb..16*nb+15).
    v2f bfrag[4][4];
#pragma unroll
    for (int j = 0; j < 4; ++j) {
        const int k0 = 4 * j + 2 * half;
#pragma unroll
        for (int nb = 0; nb < 4; ++nb) {
            const int col = nb * 16 + m;
            bfrag[j][nb].x = W[k0 * DIM + col];
            bfrag[j][nb].y = W[(k0 + 1) * DIM + col];
        }
    }

    const int numTiles = n >> 4;  // N is a multiple of 16 (N = 4194304)
    const int waveId   = (int)((blockIdx.x * blockDim.x + threadIdx.x) >> 5);
    const int numWaves = (int)((gridDim.x * blockDim.x) >> 5);

    for (int tile = waveId; tile < numTiles; tile += numWaves) {
        const int base = tile << 4;

        // Prefetch next tile's dist cacheline (speculative, dropped if OOB).
        __builtin_prefetch(dist + base + numWaves * 16, 0, 1);

        // Lanes 16-31 redundantly process the same element as lanes 0-15:
        // avoids any cross-lane shuffle when building the A fragments.
        const float d = dist[base + m];

        // searchsorted(slots, d, side='right'): count of slots[k] <= d.
        int hb = 0;
#pragma unroll
        for (int k = 0; k < N_SLOTS; ++k) hb += (s[k] <= d) ? 1 : 0;

        const bool at_end = (hb == N_SLOTS);
        int lb = hb - 1;
        lb = lb < 0 ? 0 : (lb > N_SLOTS - 1 ? N_SLOTS - 1 : lb);
        const int hbs = at_end ? lb : hb;

        const float lo    = s[lb];
        const float hi    = s[hbs];
        const float total = at_end ? 1.0f : (hi - lo);
        const float inv   = 1.0f / total;
        const float ld    = at_end ? 1.0f : (d - lo) * inv;  // scales W[hb]
        const float hd    = at_end ? 0.0f : (hi - d) * inv;  // scales W[lb]

        v8f acc[4];
#pragma unroll
        for (int nb = 0; nb < 4; ++nb)
#pragma unroll
            for (int r = 0; r < 8; ++r) acc[nb][r] = 0.0f;

        // D(16x64) = A(16x16) * W(16x64), K split into 4 chunks of 4.
#pragma unroll
        for (int j = 0; j < 4; ++j) {
            const int k0 = 4 * j + 2 * half;
            const int k1 = k0 + 1;
            v2f a;
            a.x = (k0 == lb ? hd : 0.0f) + (k0 == hbs ? ld : 0.0f);
            a.y = (k1 == lb ? hd : 0.0f) + (k1 == hbs ? ld : 0.0f);
#pragma unroll
            for (int nb = 0; nb < 4; ++nb) {
                acc[nb] = __builtin_amdgcn_wmma_f32_16x16x4_f32(
                    /*neg_a=*/false, a,
                    /*neg_b=*/false, bfrag[j][nb],
                    /*c_mod=*/(short)0, acc[nb],
                    /*reuse_a=*/false, /*reuse_b=*/false);
            }
        }

        // Scatter D: vgpr r -> row (base + r + 8*half), col nb*16 + m.
        // Each b32 store covers two contiguous 64B runs (one per lane half);
        // all 4 nb blocks + 8 r rows fill full 256B rows for L2 write combine.
#pragma unroll
        for (int nb = 0; nb < 4; ++nb) {
            float* orow = out + (size_t)(base + 8 * half) * DIM + nb * 16 + m;
#pragma unroll
            for (int r = 0; r < 8; ++r) {
                orow[(size_t)r * DIM] = acc[nb][r];
            }
        }
    }
}

extern "C" void kernel_launch(void* const* d_in, const int* in_sizes, int n_in,
                              void* d_out, int out_size, void* d_ws, size_t ws_size,
                              hipStream_t stream) {
    const float* dist  = (const float*)d_in[0];
    const float* W     = (const float*)d_in[1];  // [16][64]
    const float* slots = (const float*)d_in[2];  // [16]
    float* out = (float*)d_out;                  // [N][64]
    const int n = in_sizes[0];                   // 4194304 (multiple of 16)

    // 2048 blocks x 8 waves = 16384 waves; 262144 tiles -> 16 tiles/wave.
    dim3 grid(2048), block(256);
    dist_encoder_wmma_kernel<<<grid, block, 0, stream>>>(dist, W, slots, out, n);
}